// SSD_5068061409643
// MI455X (gfx1250) — compile-verified
//
#include <hip/hip_runtime.h>
#include <hip/hip_bf16.h>
#include <stdint.h>

#define TOPK      200
#define SORTN     256
#define NMAX      8732
#define NMS_THR   0.45f
#define CONF_THR  0.01f

// ---- CDNA5-specific helpers -------------------------------------------------

// LDS byte offset of a generic pointer into shared memory (AS0 -> AS3 cast).
__device__ __forceinline__ unsigned lds_byte_offset(const void* p) {
  return (unsigned)(unsigned long long)((__attribute__((address_space(3))) const char*)p);
}

// Async DMA: 16 bytes global -> LDS per active lane (tracked by ASYNCcnt).
__device__ __forceinline__ void async_copy_b128_to_lds(unsigned lds_off,
                                                       unsigned voff,
                                                       unsigned long long sbase) {
  asm volatile("global_load_async_to_lds_b128 %0, %1, %2 offset:0"
               :: "v"(lds_off), "v"(voff), "s"(sbase)
               : "memory");
}

__device__ __forceinline__ void wait_asynccnt0() {
#if __has_builtin(__builtin_amdgcn_s_wait_asynccnt)
  __builtin_amdgcn_s_wait_asynccnt(0);
#else
  asm volatile("s_wait_asynccnt 0x0" ::: "memory");
#endif
}

// ---- Kernel 1: box decode (2D grid: y = batch, x = anchors; no int div) -----

__global__ __launch_bounds__(256)
void ssd_decode_kernel(const float* __restrict__ loc, const float* __restrict__ dbox,
                       float* __restrict__ out, int N) {
  const int n = blockIdx.x * 256 + threadIdx.x;
  if (n >= N) return;
  const size_t i = (size_t)blockIdx.y * N + n;
  const float4 l = ((const float4*)loc)[i];
  const float4 d = ((const float4*)dbox)[n];
  float cx = d.x + l.x * 0.1f * d.z;
  float cy = d.y + l.y * 0.1f * d.w;
  float w  = d.z * expf(l.z * 0.2f);
  float h  = d.w * expf(l.w * 0.2f);
  float x1 = cx - 0.5f * w;
  float y1 = cy - 0.5f * h;
  float4 r; r.x = x1; r.y = y1; r.z = x1 + w; r.w = y1 + h;
  ((float4*)out)[i] = r;
}

// ---- Kernel 2: fused softmax + top-k (radix select) + NMS -------------------
// 2D grid: x = class (fastest -> all classes of a batch co-scheduled, conf slice
// stays L2-resident), y = batch.

__global__ __launch_bounds__(256)
void ssd_nms_kernel(const float* __restrict__ conf, const float* __restrict__ decoded,
                    float* __restrict__ out, int N, int C) {
  const int tid = threadIdx.x;
  const int c   = blockIdx.x;
  const int b   = blockIdx.y;
  float* outBlk = out + ((size_t)b * C + c) * TOPK * 5;

  if (c == 0) {  // reference multiplies class 0 by zero mask
    for (int k = tid; k < TOPK * 5; k += 256) outBlk[k] = 0.0f;
    return;
  }

  __shared__ unsigned sKey[NMAX];
  __shared__ unsigned sHist[256];
  __shared__ unsigned sListKey[SORTN];
  __shared__ int      sListIdx[SORTN];
  __shared__ int      sTieIdx[256];
  __shared__ float4   sBox[TOPK];
  __shared__ float    sArea[TOPK];
  __shared__ int      sKept[TOPK];
  __shared__ int      sPos[TOPK];
  __shared__ unsigned sPrefix;
  __shared__ int      sNeed, sCntGt, sCntEq, sNumKept;

  // --- Stage A: per-anchor softmax prob of class c -> monotonic u32 key ------
  const float* rowBase = conf + (size_t)b * N * C;
  for (int n = tid; n < N; n += 256) {
    const float* r = rowBase + (size_t)n * C;
    __builtin_prefetch((const void*)(r + (size_t)256 * C), 0, 1);  // global_prefetch_b8
    float m = r[0];
    for (int j = 1; j < C; ++j) m = fmaxf(m, r[j]);
    float s = 0.0f, ec = 0.0f;
    for (int j = 0; j < C; ++j) {
      float e = expf(r[j] - m);
      s += e;
      if (j == c) ec = e;
    }
    float p = ec / s;
    unsigned key = 0u;                         // invalid (score <= CONF_THR -> NEG)
    if (p > CONF_THR) key = __float_as_uint(p) | 0x80000000u;  // p>0 -> order-preserving
    sKey[n] = key;
  }
  if (tid == 0) { sPrefix = 0u; sNeed = TOPK; }
  __syncthreads();

  // --- Stage B: exact top-200 pivot via MSB-first radix select ---------------
  for (int round = 0; round < 4; ++round) {
    const int shift = 24 - 8 * round;
    sHist[tid] = 0u;
    __syncthreads();
    const unsigned pfx = sPrefix;
    for (int n = tid; n < N; n += 256) {
      unsigned k = sKey[n];
      if (round == 0 || (k >> (shift + 8)) == pfx)
        atomicAdd(&sHist[(k >> shift) & 255u], 1u);
    }
    __syncthreads();
    if (tid == 0) {
      int need = sNeed;
      unsigned cum = 0u;
      int chosen = 0;
      for (int bin = 255; bin >= 0; --bin) {
        unsigned h = sHist[bin];
        if (cum + h >= (unsigned)need) { chosen = bin; need -= (int)cum; break; }
        cum += h;
      }
      sPrefix = (sPrefix << 8) | (unsigned)chosen;
      sNeed   = need;
    }
    __syncthreads();
  }
  const unsigned P  = sPrefix;   // pivot key
  const int      K3 = sNeed;     // how many == P to take (ties by lowest index)

  if (tid == 0) { sCntGt = 0; sCntEq = 0; }
  __syncthreads();
  for (int n = tid; n < N; n += 256) {
    unsigned k = sKey[n];
    if (k > P) {
      int pos = atomicAdd(&sCntGt, 1);
      sListKey[pos] = k; sListIdx[pos] = n;
    }
  }
  __syncthreads();
  const int cntGt = sCntGt;      // == TOPK - K3 by radix-select invariant

  if (P != 0u) {
    for (int n = tid; n < N; n += 256) {
      if (sKey[n] == P) {
        int pos = atomicAdd(&sCntEq, 1);
        if (pos < 256) sTieIdx[pos] = n;
      }
    }
    __syncthreads();
    if (tid == 0) {
      int nt = sCntEq; if (nt > 256) nt = 256;
      for (int a = 1; a < nt; ++a) {          // ascending index order
        int v = sTieIdx[a]; int q = a - 1;
        while (q >= 0 && sTieIdx[q] > v) { sTieIdx[q + 1] = sTieIdx[q]; --q; }
        sTieIdx[q + 1] = v;
      }
      for (int t = 0; t < K3; ++t) {
        if (t < nt) { sListKey[cntGt + t] = P;  sListIdx[cntGt + t] = sTieIdx[t]; }
        else        { sListKey[cntGt + t] = 0u; sListIdx[cntGt + t] = 0x7FFFFFFF; }
      }
    }
  } else {
    for (int t = cntGt + tid; t < TOPK; t += 256) {  // fewer than 200 valid
      sListKey[t] = 0u; sListIdx[t] = 0x7FFFFFFF;
    }
  }
  if (tid >= TOPK) { sListKey[tid] = 0u; sListIdx[tid] = 0x7FFFFFFF; }  // pad to 256
  __syncthreads();

  // --- Stage C: bitonic sort 256 (key desc, idx asc) -------------------------
  for (int kk = 2; kk <= SORTN; kk <<= 1) {
    for (int j = kk >> 1; j > 0; j >>= 1) {
      int ixj = tid ^ j;
      if (ixj > tid) {
        unsigned ka = sListKey[tid], kb = sListKey[ixj];
        int ia = sListIdx[tid], ib = sListIdx[ixj];
        bool before = (ka > kb) || (ka == kb && ia < ib);
        bool doSwap = ((tid & kk) == 0) ? !before : before;
        if (doSwap) {
          sListKey[tid] = kb; sListKey[ixj] = ka;
          sListIdx[tid] = ib; sListIdx[ixj] = ia;
        }
      }
      __syncthreads();
    }
  }

  // --- Stage D: async-DMA gather of the 200 selected boxes into LDS ----------
  const unsigned long long boxBase =
      (unsigned long long)(uintptr_t)(decoded + (size_t)b * N * 4);
  if (tid < TOPK) {
    if (sListKey[tid] != 0u) {
      async_copy_b128_to_lds(lds_byte_offset(&sBox[tid]),
                             (unsigned)sListIdx[tid] * 16u, boxBase);
    } else {
      sBox[tid] = make_float4(0.f, 0.f, 0.f, 0.f);
    }
  }
  wait_asynccnt0();
  __syncthreads();
  if (tid < TOPK) {
    float4 bb = sBox[tid];
    sArea[tid] = (bb.z - bb.x) * (bb.w - bb.y);
    sKept[tid] = (sListKey[tid] != 0u) ? 1 : 0;
  }
  __syncthreads();

  // --- Stage E: greedy NMS (one pair-column per lane, barrier per pivot) -----
  for (int i = 0; i < TOPK - 1; ++i) {
    if (sKept[i]) {
      int j = tid;
      if (j > i && j < TOPK && sKept[j]) {
        float4 a = sBox[i], bb = sBox[j];
        float iw = fmaxf(fminf(a.z, bb.z) - fmaxf(a.x, bb.x), 0.f);
        float ih = fmaxf(fminf(a.w, bb.w) - fmaxf(a.y, bb.y), 0.f);
        float inter = iw * ih;
        float uni = sArea[i] + sArea[j] - inter;
        if (inter / uni > NMS_THR) sKept[j] = 0;
      }
    }
    __syncthreads();
  }

  // --- Stage F: compact kept (desc score order), zero remainder --------------
  if (tid == 0) {
    int p = 0;
    for (int t = 0; t < TOPK; ++t) { sPos[t] = sKept[t] ? p : -1; p += sKept[t]; }
    sNumKept = p;
  }
  __syncthreads();
  if (tid < TOPK) {
    if (sKept[tid]) {
      float sc = __uint_as_float(sListKey[tid] & 0x7FFFFFFFu);
      float4 bb = sBox[tid];
      float* o = outBlk + (size_t)sPos[tid] * 5;
      o[0] = sc; o[1] = bb.x; o[2] = bb.y; o[3] = bb.z; o[4] = bb.w;
    }
    if (tid >= sNumKept) {
      float* o = outBlk + (size_t)tid * 5;
      o[0] = 0.f; o[1] = 0.f; o[2] = 0.f; o[3] = 0.f; o[4] = 0.f;
    }
  }
}

// ---- Launcher ---------------------------------------------------------------

extern "C" void kernel_launch(void* const* d_in, const int* in_sizes, int n_in,
                              void* d_out, int out_size, void* d_ws, size_t ws_size,
                              hipStream_t stream) {
  (void)n_in; (void)out_size; (void)ws_size;
  const float* loc  = (const float*)d_in[0];   // [B, N, 4]
  const float* conf = (const float*)d_in[1];   // [B, N, C]
  const float* dbox = (const float*)d_in[2];   // [N, 4]
  const int N = in_sizes[2] / 4;
  const int B = in_sizes[0] / (4 * N);
  const int C = in_sizes[1] / (B * N);

  float* decoded = (float*)d_ws;               // [B, N, 4] scratch (~18 MB)
  ssd_decode_kernel<<<dim3((N + 255) / 256, B), 256, 0, stream>>>(loc, dbox, decoded, N);
  ssd_nms_kernel<<<dim3(C, B), 256, 0, stream>>>(conf, decoded, (float*)d_out, N, C);
}